// HeadlessAttention_26766236189059
// MI455X (gfx1250) — compile-verified
//
#include <hip/hip_runtime.h>
#include <math.h>

typedef __attribute__((ext_vector_type(2))) float v2f;
typedef __attribute__((ext_vector_type(8))) float v8f;

#define T_DIM 128
#define Q_DIM 512
#define K_DIM 512
#define B_DIM 2

// ---------------------------------------------------------------------------
// Generic linear: C[m][u] = sum_t X[m][t] * W[u][t]  (x @ W.T), T fixed = 128.
// One 16x16 output tile per wave, fp32 WMMA 16x16x4, K-loop of 32 steps.
// A-frag: lane l holds X[M0 + (l&15)][4kk + 2*(l>>4) + {0,1}]   (f32 A 16x4 layout)
// B-frag: lane l holds W[N0 + (l&15)][4kk + 2*(l>>4) + {0,1}]   (B[t][u] = W[u][t])
// ---------------------------------------------------------------------------
__global__ void gemm_xwT_kernel(const float* __restrict__ X,
                                const float* __restrict__ W,
                                float* __restrict__ C, int M) {
  const int lane = threadIdx.x & 31;
  const int wave = threadIdx.x >> 5;
  const int tile = blockIdx.x * 8 + wave;
  const int ntiles = (M >> 4) * (T_DIM >> 4);
  if (tile >= ntiles) return;
  const int mt = tile >> 3;          // T_DIM/16 == 8 N-tiles
  const int nt = tile & 7;
  const int lr = lane & 15;
  const int lh = lane >> 4;

  const float* xrow = X + (mt * 16 + lr) * T_DIM + 2 * lh;
  const float* wrow = W + (nt * 16 + lr) * T_DIM + 2 * lh;

  v8f acc = {};
#pragma unroll
  for (int kk = 0; kk < 32; ++kk) {
    v2f a = *(const v2f*)(xrow + 4 * kk);
    v2f b = *(const v2f*)(wrow + 4 * kk);
    acc = __builtin_amdgcn_wmma_f32_16x16x4_f32(false, a, false, b,
                                                (short)0, acc, false, false);
  }

  // C layout: vgpr r, lane l -> row M0 + r + 8*(l>>4), col N0 + (l&15)
  float* crow = C + (mt * 16 + 8 * lh) * T_DIM + nt * 16 + lr;
#pragma unroll
  for (int r = 0; r < 8; ++r) crow[r * T_DIM] = acc[r];
}

__global__ void zero_kernel(float* __restrict__ p, int n) {
  int i = blockIdx.x * blockDim.x + threadIdx.x;
  if (i < n) p[i] = 0.0f;
}

// ---------------------------------------------------------------------------
// Per-(b,k) fused kernel. 256 threads (8 wave32). Dynamic LDS = 320 KB:
//   smem[0      .. 65536)  : L[q][u]  (512 x 128 f32 logits tile, stride 128)
//   smem[65536  .. 81920)  : Wlk in exact B-fragment order (64 KB);
//                            reused after the GEMM as reduction scratch.
// Steps:
//   1) Wlk[u][t] = Wl[u][t] * k[b,k,t], scattered into B-frag order
//   2) L = q_mat(b) @ Wlk.T via f32 WMMA (8192 wmma / WG)
//   3) swishmax over Q per column u:  m = max_q L; e = L*exp(L-m);
//      denom = sum_q |e| + 1;  coef[u] = v[b,k,u] / denom
//   4) atomicAdd vsum[b,q,u] += coef[u] * e[q,u]
// ---------------------------------------------------------------------------
#define L_FLOATS (Q_DIM * T_DIM)      /* 65536 */
#define BF_OFF   L_FLOATS
#define RED_MAX  BF_OFF
#define RED_SUM  (BF_OFF + 256)
#define COEF_OFF (BF_OFF + 512)

__global__ void attn_bk_kernel(const float* __restrict__ q_pre,
                               const float* __restrict__ k_pre,
                               const float* __restrict__ v_pre,
                               const float* __restrict__ Wl,
                               float* __restrict__ vsum) {
  extern __shared__ float smem[];
  const int kblk = blockIdx.x;       // 0..511
  const int b    = blockIdx.y;       // 0..1
  const int tid  = threadIdx.x;      // 0..255

  const float* kvec = k_pre + (b * K_DIM + kblk) * T_DIM;

  // ---- Stage 1: fold k into Wl, store in per-lane B-fragment order --------
  for (int i = tid; i < T_DIM * T_DIM; i += 256) {
    const int u = i >> 7, t = i & 127;
    const float w = Wl[i] * kvec[t];
    const int n = u >> 4, lr = u & 15;
    const int kk = t >> 2, rem = t & 3;
    const int lane = lr + ((rem >> 1) << 4);
    // index = ((n*32 + kk)*32 + lane)*2 + (rem&1)
    smem[BF_OFF + ((((n << 5) + kk) << 5) + lane) * 2 + (rem & 1)] = w;
  }
  __syncthreads();

  // ---- Stage 2: L[q][u] = sum_t q_pre[b][q][t] * Wlk[u][t] ----------------
  {
    const int lane = tid & 31;
    const int wave = tid >> 5;
    const int lr = lane & 15;
    const int lh = lane >> 4;
    for (int mt = wave; mt < 32; mt += 8) {     // 32 M-tiles of 16 rows
      const float* qrow = q_pre + ((b * Q_DIM) + mt * 16 + lr) * T_DIM + 2 * lh;
      v2f a[32];
#pragma unroll
      for (int kk = 0; kk < 32; ++kk) a[kk] = *(const v2f*)(qrow + 4 * kk);

      for (int nt = 0; nt < 8; ++nt) {
        const float* bf = smem + BF_OFF + (nt << 11) + (lane << 1);
        v8f acc = {};
#pragma unroll
        for (int kk = 0; kk < 32; ++kk) {
          v2f bb = *(const v2f*)(bf + (kk << 6));   // conflict-free ds b64
          acc = __builtin_amdgcn_wmma_f32_16x16x4_f32(false, a[kk], false, bb,
                                                      (short)0, acc, false, false);
        }
        float* lrow = smem + (mt * 16 + 8 * lh) * T_DIM + nt * 16 + lr;
#pragma unroll
        for (int r = 0; r < 8; ++r) lrow[r * T_DIM] = acc[r];
      }
    }
  }
  __syncthreads();

  // ---- Stage 3: swishmax over Q, per column u -----------------------------
  const int u = tid & 127;
  const int h = tid >> 7;            // each (u, half) owns 256 q-rows
  {
    float m = -INFINITY;
    for (int q = h * 256; q < h * 256 + 256; ++q)
      m = fmaxf(m, smem[q * T_DIM + u]);
    smem[RED_MAX + tid] = m;         // scratch reuses B-frag region (GEMM done)
  }
  __syncthreads();
  {
    const float m = fmaxf(smem[RED_MAX + u], smem[RED_MAX + 128 + u]);
    float s = 0.0f;
    for (int q = h * 256; q < h * 256 + 256; ++q) {
      const float x = smem[q * T_DIM + u];
      const float e = x * __expf(x - m);
      smem[q * T_DIM + u] = e;       // overwrite L with xexp
      s += fabsf(e);
    }
    smem[RED_SUM + tid] = s;
  }
  __syncthreads();
  if (tid < 128) {
    const float denom = smem[RED_SUM + tid] + smem[RED_SUM + 128 + tid] + 1.0f;  // NOT_EPSILON
    smem[COEF_OFF + tid] = v_pre[(b * K_DIM + kblk) * T_DIM + tid] / denom;
  }
  __syncthreads();

  // ---- Stage 4: vsum[b,q,u] += coef[u] * xexp[q,u]  (L2-resident atomics) --
  float* vs = vsum + b * Q_DIM * T_DIM;
  for (int i = tid; i < Q_DIM * T_DIM; i += 256) {
    const float val = smem[i] * smem[COEF_OFF + (i & 127)];
    atomicAdd(vs + i, val);
  }
}

// ---------------------------------------------------------------------------
extern "C" void kernel_launch(void* const* d_in, const int* in_sizes, int n_in,
                              void* d_out, int out_size, void* d_ws, size_t ws_size,
                              hipStream_t stream) {
  const float* query = (const float*)d_in[0];
  const float* key   = (const float*)d_in[1];
  const float* value = (const float*)d_in[2];
  const float* Wk    = (const float*)d_in[3];
  const float* Wq    = (const float*)d_in[4];
  const float* Wv    = (const float*)d_in[5];
  const float* Wl    = (const float*)d_in[6];
  const float* Wo    = (const float*)d_in[7];
  float* out = (float*)d_out;

  float* ws    = (float*)d_ws;
  float* k_pre = ws;                 // [B*K, T]  131072 f
  float* q_pre = ws + 131072;        // [B*Q, T]
  float* v_pre = ws + 262144;        // [B*K, T]
  float* vsum  = ws + 393216;        // [B*Q, T]

  const int M = B_DIM * K_DIM;       // 1024 rows for every linear here
  const int gemm_blocks = ((M >> 4) * (T_DIM >> 4) + 7) / 8;   // 64

  zero_kernel<<<dim3((B_DIM * Q_DIM * T_DIM) / 256), 256, 0, stream>>>(vsum, B_DIM * Q_DIM * T_DIM);

  gemm_xwT_kernel<<<dim3(gemm_blocks), 256, 0, stream>>>(key,   Wk, k_pre, M);
  gemm_xwT_kernel<<<dim3(gemm_blocks), 256, 0, stream>>>(query, Wq, q_pre, M);
  gemm_xwT_kernel<<<dim3(gemm_blocks), 256, 0, stream>>>(value, Wv, v_pre, M);

  // One workgroup per (b,k); 320 KB dynamic LDS (CDNA5 max per workgroup).
  attn_bk_kernel<<<dim3(K_DIM, B_DIM), 256, 320 * 1024, stream>>>(q_pre, k_pre, v_pre, Wl, vsum);

  gemm_xwT_kernel<<<dim3(gemm_blocks), 256, 0, stream>>>(vsum, Wo, out, M);
}